// SpeakerEncoder_48790828483171
// MI455X (gfx1250) — compile-verified
//
#include <hip/hip_runtime.h>
#include <stdint.h>

typedef __attribute__((ext_vector_type(4))) float        v4f;
typedef __attribute__((ext_vector_type(4))) unsigned int v4u;
typedef __attribute__((ext_vector_type(4))) int          v4i;
typedef __attribute__((ext_vector_type(8))) int          v8i;

#define L_DIM 8
#define K_DIM 1024
#define D_DIM 1024
#define NPB   8      // samples (n) per block, pipelined
#define TPB   256    // 8 waves (wave32)

#if defined(__gfx1250__) && __has_builtin(__builtin_amdgcn_tensor_load_to_lds)
#define USE_TDM 1
#else
#define USE_TDM 0
#endif

// Gathered rows are viewed as a (L*K, D) fp32 matrix; row index = l*K + x[n,l].
// TDM gather mode (32-bit indices) fetches up to 8 rows per descriptor -> exactly L=8.

__global__ __launch_bounds__(TPB) void embsum_tdm_kernel(const int* __restrict__ x,
                                                         const float* __restrict__ w,
                                                         float* __restrict__ out)
{
    __shared__ alignas(128) float lds[2 * L_DIM * D_DIM]; // 2 x 32 KB double buffer

    const int tid = threadIdx.x;
    const long long n0 = (long long)blockIdx.x * NPB;

#if USE_TDM
    const unsigned long long wga = (unsigned long long)(uintptr_t)w;
    // low 32 bits of a generic LDS pointer == byte offset within the wave's LDS
    const unsigned lds0 = (unsigned)(uintptr_t)(&lds[0]);

    // ---- D# group 0 (static part): count=1 | gather_index_size=32b | gather_mode ----
    v4u g0base;
    g0base[0] = 0xC0000001u;
    g0base[1] = 0u;                                            // lds_addr, per issue
    g0base[2] = (unsigned)(wga & 0xffffffffull);               // global_addr[31:0]
    g0base[3] = (unsigned)((wga >> 32) & 0x01ffffffull)        // global_addr[56:32]
              | 0x80000000u;                                   // type = 2 ("image")

    // ---- D# group 1: data_size=4B, tensor 8192x1024, tile 1024x8(indices) ----
    v8i g1;
    g1[0] = (int)(2u << 16);                                   // data_size=4B, wg_mask=0
    g1[1] = (int)((unsigned)(D_DIM & 0xffff) << 16);           // tensor_dim0 lo16
    g1[2] = (int)(((unsigned)D_DIM >> 16)
          | ((unsigned)((L_DIM * K_DIM) & 0xffff) << 16));     // dim0 hi16 | dim1 lo16
    g1[3] = (int)(((unsigned)(L_DIM * K_DIM) >> 16)
          | ((unsigned)D_DIM << 16));                          // dim1 hi16 | tile_dim0
    g1[4] = L_DIM;                                             // tile_dim1 = #indices
    g1[5] = D_DIM;                                             // tensor_dim0_stride lo32
    g1[6] = 0;                                                 // stride hi16 (dim1 stride ignored)
    g1[7] = 0;

    // Extra operand of the 6-arg builtin (clang-23 form): hardware defines only
    // four D# groups (VADDR4 is unused/NULL in the VIMAGE encoding) -> zero fill.
    const v8i gx = {0, 0, 0, 0, 0, 0, 0, 0};

    auto issue = [&](int i) {
        const int* xr = x + (n0 + i) * L_DIM;
        v4i g2, g3;
        #pragma unroll
        for (int l = 0; l < 4; ++l)
            g2[l] = __builtin_amdgcn_readfirstlane(l * K_DIM + xr[l]);
        #pragma unroll
        for (int l = 0; l < 4; ++l)
            g3[l] = __builtin_amdgcn_readfirstlane((l + 4) * K_DIM + xr[l + 4]);
        v4u g0 = g0base;
        g0[1] = lds0 + (unsigned)((i & 1) * (L_DIM * D_DIM * 4));
        __builtin_amdgcn_tensor_load_to_lds(g0, g1, g2, g3, gx, /*cpol=*/0);
    };

    if (tid < 32) issue(0);                                    // prologue

    for (int i = 0; i < NPB; ++i) {
        if (tid < 32) {
            if (i + 1 < NPB) {
                issue(i + 1);                                  // prefetch next sample
                __builtin_amdgcn_s_wait_tensorcnt(1);          // current gather done
            } else {
                __builtin_amdgcn_s_wait_tensorcnt(0);
            }
        }
        __syncthreads();

        const float* buf = &lds[(i & 1) * (L_DIM * D_DIM)];
        v4f acc = *(const v4f*)(buf + tid * 4);
        #pragma unroll
        for (int r = 1; r < L_DIM; ++r)
            acc += *(const v4f*)(buf + r * D_DIM + tid * 4);

        // NT store: 512 MB output stream must not evict the 32 MB L2-resident table
        __builtin_nontemporal_store(acc, (v4f*)(out + (n0 + i) * (long long)D_DIM + tid * 4));

        __syncthreads();                                       // buffer reuse fence
    }
#else
    // Fallback: direct coalesced gather (weight stays L2-resident), NT output stores
    for (int i = 0; i < NPB; ++i) {
        const long long n = n0 + i;
        const int* xr = x + n * L_DIM;
        v4f acc = {0.f, 0.f, 0.f, 0.f};
        #pragma unroll
        for (int l = 0; l < L_DIM; ++l) {
            const float* row = w + ((long long)l * K_DIM + xr[l]) * D_DIM;
            acc += *(const v4f*)(row + tid * 4);
        }
        __builtin_nontemporal_store(acc, (v4f*)(out + n * (long long)D_DIM + tid * 4));
    }
#endif
}

extern "C" void kernel_launch(void* const* d_in, const int* in_sizes, int n_in,
                              void* d_out, int out_size, void* d_ws, size_t ws_size,
                              hipStream_t stream) {
    const int*   x   = (const int*)d_in[0];     // (N, 8) int32
    const float* w   = (const float*)d_in[1];   // (8, 1024, 1024) fp32
    float*       out = (float*)d_out;           // (N, 1024) fp32

    const int N = in_sizes[0] / L_DIM;          // 131072
    const int blocks = N / NPB;                 // 16384
    embsum_tdm_kernel<<<blocks, TPB, 0, stream>>>(x, w, out);
}